// XLMRCRFForTokenClassification_70308614636103
// MI455X (gfx1250) — compile-verified
//
#include <hip/hip_runtime.h>
#include <hip/hip_bf16.h>
#include <math.h>

#define Bb 64
#define Tt 512
#define Hh 768
#define Cc 21
#define Mrows (Bb * Tt)   // 32768

typedef float v2f __attribute__((ext_vector_type(2)));
typedef float v8f __attribute__((ext_vector_type(8)));

// ---------------------------------------------------------------------------
// Kernel 1: emissions = hidden @ W + b  via V_WMMA_F32_16X16X4_F32
// 256 threads = 8 waves; each wave computes one 16-row x 21-col output tile.
// Block covers 128 rows; grid = 32768/128 = 256 blocks.
// ---------------------------------------------------------------------------
__global__ __launch_bounds__(256) void emissions_gemm_kernel(
    const float* __restrict__ hidden,  // [Mrows, Hh]
    const float* __restrict__ W,       // [Hh, Cc]
    const float* __restrict__ bias,    // [Cc]
    float* __restrict__ em)            // [Mrows, Cc]
{
    __shared__ float w_lds[Hh * Cc];   // 64512 bytes
    const int tid = threadIdx.x;
    for (int i = tid; i < Hh * Cc; i += 256) w_lds[i] = W[i];
    __syncthreads();

    const int wave = tid >> 5;
    const int lane = tid & 31;
    const int row0 = blockIdx.x * 128 + wave * 16;
    const int lrow = lane & 15;    // A row / D column within tile
    const int lhi  = lane >> 4;    // 0: K pair {0,1}, 1: K pair {2,3}
    const int kk   = 2 * lhi;

    // A pointer: lane l loads hidden[row0+lrow][k + kk .. k+kk+1] as b64
    const float* Aptr = hidden + (size_t)(row0 + lrow) * Hh + kk;

    // B columns for the two N-tiles (clamp tile1 cols >=21 to a valid address;
    // those D columns are never stored, so their values are irrelevant).
    const int nb0 = lrow;                               // 0..15, always valid
    const int nb1 = (16 + lrow < Cc) ? (16 + lrow) : (Cc - 1);

    v8f acc0 = {};
    v8f acc1 = {};

#pragma unroll 4
    for (int k = 0; k < Hh; k += 4) {
        v2f a = *(const v2f*)(Aptr + k);
        // B layout (32-bit 4x16): VGPR0 = row (k+kk), VGPR1 = row (k+kk+1),
        // column = lane&15 (lanes 16-31 carry K=2,3 rows).
        v2f b0, b1;
        b0.x = w_lds[(k + kk) * Cc + nb0];
        b0.y = w_lds[(k + kk + 1) * Cc + nb0];
        b1.x = w_lds[(k + kk) * Cc + nb1];
        b1.y = w_lds[(k + kk + 1) * Cc + nb1];
        acc0 = __builtin_amdgcn_wmma_f32_16x16x4_f32(false, a, false, b0,
                                                     (short)0, acc0, false, false);
        acc1 = __builtin_amdgcn_wmma_f32_16x16x4_f32(false, a, false, b1,
                                                     (short)0, acc1, false, false);
    }

    const float bv0 = bias[nb0];
    const bool  t1v = (16 + lrow) < Cc;
    const float bv1 = t1v ? bias[16 + lrow] : 0.0f;

    // D layout: VGPR i holds M=i (lanes 0-15) or M=8+i (lanes 16-31), N = lane&15.
#pragma unroll
    for (int i = 0; i < 8; ++i) {
        const int r = row0 + i + 8 * lhi;
        em[(size_t)r * Cc + lrow] = acc0[i] + bv0;
        if (t1v) em[(size_t)r * Cc + 16 + lrow] = acc1[i] + bv1;
    }
}

// ---------------------------------------------------------------------------
// Kernel 2: CRF negative log-likelihood pieces, one wave (32 lanes) per batch.
// Lane c (c < 21) owns alpha[c]; cross-lane broadcast via __shfl (wave32).
// ---------------------------------------------------------------------------
__global__ __launch_bounds__(32) void crf_kernel(
    const float* __restrict__ em_all,   // [Bb, Tt, Cc]
    const int*   __restrict__ labels,   // [Bb, Tt]  (-100 = padding)
    const float* __restrict__ startt,   // [Cc]
    const float* __restrict__ endt,     // [Cc]
    const float* __restrict__ trans,    // [Cc, Cc]
    float* __restrict__ llh_out)        // [Bb]
{
    const int b    = blockIdx.x;
    const int lane = threadIdx.x;
    const float* em  = em_all + (size_t)b * Tt * Cc;
    const int*   lab = labels + (size_t)b * Tt;

    // ---- numerator: start + sum(em[t,tag_t]) + sum(trans[tag_{t-1},tag_t]) + end
    float partial = 0.0f;
    int   cnt     = 0;
    for (int t = lane; t < Tt; t += 32) {
        const int tg = lab[t];
        if (tg != -100) {
            ++cnt;
            float v = em[t * Cc + tg];
            if (t > 0) v += trans[lab[t - 1] * Cc + tg];
            partial += v;
        }
    }
#pragma unroll
    for (int off = 16; off > 0; off >>= 1) {
        partial += __shfl_xor(partial, off);
        cnt     += __shfl_xor(cnt, off);
    }
    float num = partial + startt[lab[0]] + endt[lab[cnt - 1]];

    // ---- forward recurrence (mask is a prefix: valid t in [0, cnt) )
    const int  c  = lane;
    const bool cv = (c < Cc);
    float trcol[Cc];
#pragma unroll
    for (int p = 0; p < Cc; ++p) trcol[p] = cv ? trans[p * Cc + c] : 0.0f;

    float alpha = cv ? (startt[c] + em[c]) : -INFINITY;

    for (int t = 1; t < cnt; ++t) {
        float v[Cc];
        float m = -INFINITY;
#pragma unroll
        for (int p = 0; p < Cc; ++p) {
            const float ap = __shfl(alpha, p);
            v[p] = ap + trcol[p];
            m = fmaxf(m, v[p]);
        }
        float s = 0.0f;
#pragma unroll
        for (int p = 0; p < Cc; ++p) s += __expf(v[p] - m);
        const float e = cv ? em[t * Cc + c] : 0.0f;
        alpha = m + __logf(s) + e;
    }

    // ---- logZ = logsumexp(alpha + end)
    const float ec = cv ? endt[c] : -INFINITY;
    float va = alpha + ec;
    float m = va;
#pragma unroll
    for (int off = 16; off > 0; off >>= 1) m = fmaxf(m, __shfl_xor(m, off));
    float s = __expf(va - m);
#pragma unroll
    for (int off = 16; off > 0; off >>= 1) s += __shfl_xor(s, off);
    const float logZ = m + __logf(s);

    if (lane == 0) llh_out[b] = num - logZ;
}

// ---------------------------------------------------------------------------
// Kernel 3: loss = -mean(llh)
// ---------------------------------------------------------------------------
__global__ void reduce_kernel(const float* __restrict__ llh, float* __restrict__ out)
{
    float s = 0.0f;
    for (int i = 0; i < Bb; ++i) s += llh[i];
    out[0] = -s / (float)Bb;
}

extern "C" void kernel_launch(void* const* d_in, const int* in_sizes, int n_in,
                              void* d_out, int out_size, void* d_ws, size_t ws_size,
                              hipStream_t stream)
{
    const float* hidden = (const float*)d_in[0];
    const float* W      = (const float*)d_in[1];
    const float* bias   = (const float*)d_in[2];
    const float* startt = (const float*)d_in[3];
    const float* endt   = (const float*)d_in[4];
    const float* trans  = (const float*)d_in[5];
    const int*   labels = (const int*)d_in[6];

    float* em  = (float*)d_ws;                        // Mrows*Cc floats
    float* llh = em + (size_t)Mrows * Cc;             // Bb floats

    emissions_gemm_kernel<<<Mrows / 128, 256, 0, stream>>>(hidden, W, bias, em);
    crf_kernel<<<Bb, 32, 0, stream>>>(em, labels, startt, endt, trans, llh);
    reduce_kernel<<<1, 1, 0, stream>>>(llh, (float*)d_out);
}